// UnfeatGraphIsomNet_24154896073106
// MI455X (gfx1250) — compile-verified
//
#include <hip/hip_runtime.h>
#include <hip/hip_bf16.h>

typedef __attribute__((ext_vector_type(2))) float v2f;
typedef __attribute__((ext_vector_type(8))) float v8f;

#define DIM 64
#define NCLS 16

// ---------------------------------------------------------------------------
// Edge scatter-add: agg[dst] += h[src].  One wave32 per edge; lane l moves
// float2 at feature offset 2*l (32 lanes * 2 floats = 64 features).
// Feature matrix (25.6MB) lives in the 192MB L2, so these are L2 atomics.
// ---------------------------------------------------------------------------
__global__ __launch_bounds__(256) void gin_agg_kernel(
    const float* __restrict__ h, const int* __restrict__ src,
    const int* __restrict__ dst, float* __restrict__ agg, int nEdges) {
  const int lane = threadIdx.x & 31;
  const int wave = blockIdx.x * (blockDim.x >> 5) + (threadIdx.x >> 5);
  const int nWaves = gridDim.x * (blockDim.x >> 5);
  for (int e = wave; e < nEdges; e += nWaves) {
    const int s = src[e];
    const int d = dst[e];
    const float2 v = ((const float2*)(h + (size_t)s * DIM))[lane];
    float* ap = agg + (size_t)d * DIM + lane * 2;
    atomicAdd(ap, v.x);      // global_atomic_add_f32
    atomicAdd(ap + 1, v.y);  // global_atomic_add_f32
  }
}

// ---------------------------------------------------------------------------
// One GIN layer GEMM: hout[16 rows] = relu((hin + agg) @ W), W is [64][64]
// row-major ([K][N]).  One wave per 16-node tile; 4 C-tiles of 16x16 f32,
// K-loop of 16 steps of v_wmma_f32_16x16x4_f32 (full fp32 precision).
//
// f32 WMMA fragment layout (ISA 7.12.2):
//   A 16x4 : lane m=l&15 holds row m; VGPR0/1 = K = 2*(l>>4), 2*(l>>4)+1
//   B 4x16 : lane n=l&15 holds col n; VGPR0/1 = K = 2*(l>>4), 2*(l>>4)+1
//   C 16x16: VGPR r, lanes 0-15 -> (M=r, N=l); lanes 16-31 -> (M=r+8, N=l-16)
// In-place (hout==hin) is safe: each wave reads only its own 16 rows into
// registers before any store.
// ---------------------------------------------------------------------------
__global__ __launch_bounds__(256) void gin_gemm_kernel(
    const float* __restrict__ hin, const float* __restrict__ agg,
    const float* __restrict__ W, float* __restrict__ hout, int nTiles) {
  const int lane = threadIdx.x & 31;
  const int tile = blockIdx.x * (blockDim.x >> 5) + (threadIdx.x >> 5);
  if (tile >= nTiles) return;  // wave-uniform: EXEC stays all-ones for WMMA

  const int m = lane & 15;
  const int hi = lane >> 4;
  const int kb = hi * 2;
  const int row = tile * 16 + m;

  const float* hrow = hin + (size_t)row * DIM;
  const float* arow = agg + (size_t)row * DIM;

  // A fragments for all 16 K-steps: (h + agg) row slice for this lane.
  v2f afrag[16];
#pragma unroll
  for (int k = 0; k < 16; ++k) {
    const int kk = 4 * k + kb;
    v2f a;
    a.x = hrow[kk] + arow[kk];
    a.y = hrow[kk + 1] + arow[kk + 1];
    afrag[k] = a;
  }

#pragma unroll
  for (int nt = 0; nt < 4; ++nt) {
    const int col = nt * 16 + m;
    v8f acc = {};
#pragma unroll
    for (int k = 0; k < 16; ++k) {
      const int kk = 4 * k + kb;
      v2f b;
      b.x = W[kk * DIM + col];
      b.y = W[(kk + 1) * DIM + col];
      acc = __builtin_amdgcn_wmma_f32_16x16x4_f32(
          /*neg_a=*/false, afrag[k], /*neg_b=*/false, b,
          /*c_mod=*/(short)0, acc, /*reuse_a=*/false, /*reuse_b=*/false);
    }
#pragma unroll
    for (int r = 0; r < 8; ++r) {
      float v = acc[r];
      v = v > 0.0f ? v : 0.0f;  // relu
      hout[(size_t)(tile * 16 + r + 8 * hi) * DIM + col] = v;
    }
  }
}

// ---------------------------------------------------------------------------
// L2 row-normalize with eps clamp: feat = h / max(||h||, 1e-12).
// One wave per node; butterfly shuffle reduction over 32 lanes.
// ---------------------------------------------------------------------------
__global__ __launch_bounds__(256) void gin_norm_kernel(
    const float* __restrict__ h, float* __restrict__ feat, int nNodes) {
  const int lane = threadIdx.x & 31;
  const int wave = blockIdx.x * (blockDim.x >> 5) + (threadIdx.x >> 5);
  const int nWaves = gridDim.x * (blockDim.x >> 5);
  for (int node = wave; node < nNodes; node += nWaves) {
    const float2 v = ((const float2*)(h + (size_t)node * DIM))[lane];
    float s = v.x * v.x + v.y * v.y;
#pragma unroll
    for (int off = 16; off > 0; off >>= 1) s += __shfl_xor(s, off, 32);
    const float scale = 1.0f / fmaxf(sqrtf(s), 1e-12f);
    float2 o;
    o.x = v.x * scale;
    o.y = v.y * scale;
    ((float2*)(feat + (size_t)node * DIM))[lane] = o;
  }
}

// ---------------------------------------------------------------------------
// Output projection: out = feat @ W_out + b_out, W_out [64][16] row-major.
// One 16x16 WMMA C-tile per 16-node group.
// ---------------------------------------------------------------------------
__global__ __launch_bounds__(256) void gin_proj_kernel(
    const float* __restrict__ feat, const float* __restrict__ Wout,
    const float* __restrict__ bout, float* __restrict__ out, int nTiles) {
  const int lane = threadIdx.x & 31;
  const int tile = blockIdx.x * (blockDim.x >> 5) + (threadIdx.x >> 5);
  if (tile >= nTiles) return;

  const int m = lane & 15;
  const int hi = lane >> 4;
  const int kb = hi * 2;
  const float* frow = feat + (size_t)(tile * 16 + m) * DIM;

  v8f acc = {};
#pragma unroll
  for (int k = 0; k < 16; ++k) {
    const int kk = 4 * k + kb;
    v2f a;
    a.x = frow[kk];
    a.y = frow[kk + 1];
    v2f b;
    b.x = Wout[kk * NCLS + m];
    b.y = Wout[(kk + 1) * NCLS + m];
    acc = __builtin_amdgcn_wmma_f32_16x16x4_f32(
        false, a, false, b, (short)0, acc, false, false);
  }
  const float bias = bout[m];
#pragma unroll
  for (int r = 0; r < 8; ++r) {
    out[(size_t)(tile * 16 + r + 8 * hi) * NCLS + m] = acc[r] + bias;
  }
}

// ---------------------------------------------------------------------------
// Launch: 3 x (zero agg -> scatter -> WMMA gemm), then normalize + project.
// d_in: x, src, dst, W0, W1, W2, W_out, b_out
// d_out: out[nNodes*16] followed by feat[nNodes*64]
// d_ws: hbuf[nTiles*16*64] + agg[nTiles*16*64] (fp32)
// ---------------------------------------------------------------------------
extern "C" void kernel_launch(void* const* d_in, const int* in_sizes, int n_in,
                              void* d_out, int out_size, void* d_ws,
                              size_t ws_size, hipStream_t stream) {
  const float* x = (const float*)d_in[0];
  const int* src = (const int*)d_in[1];
  const int* dst = (const int*)d_in[2];
  const float* Ws[3] = {(const float*)d_in[3], (const float*)d_in[4],
                        (const float*)d_in[5]};
  const float* Wout = (const float*)d_in[6];
  const float* bout = (const float*)d_in[7];

  const int nNodes = in_sizes[0] / DIM;
  const int nEdges = in_sizes[1];
  const int nTiles = (nNodes + 15) / 16;

  float* outp = (float*)d_out;                 // [nNodes * 16]
  float* featp = outp + (size_t)nNodes * NCLS; // [nNodes * 64]

  float* hbuf = (float*)d_ws;                          // padded to tile mult.
  float* agg = hbuf + (size_t)nTiles * 16 * DIM;
  const size_t aggBytes = (size_t)nTiles * 16 * DIM * sizeof(float);

  const int wpb = 8;  // 256 threads = 8 wave32
  const int gemmBlocks = (nTiles + wpb - 1) / wpb;
  const int aggBlocks = 2048;   // grid-stride over edges
  const int normBlocks = 2048;  // grid-stride over nodes

  const float* hcur = x;
  for (int l = 0; l < 3; ++l) {
    hipMemsetAsync(agg, 0, aggBytes, stream);
    gin_agg_kernel<<<aggBlocks, 256, 0, stream>>>(hcur, src, dst, agg, nEdges);
    gin_gemm_kernel<<<gemmBlocks, 256, 0, stream>>>(hcur, agg, Ws[l], hbuf,
                                                    nTiles);
    hcur = hbuf;  // layers 1,2 update hbuf in place
  }

  gin_norm_kernel<<<normBlocks, 256, 0, stream>>>(hbuf, featp, nNodes);
  gin_proj_kernel<<<gemmBlocks, 256, 0, stream>>>(featp, Wout, bout, outp,
                                                  nTiles);
}